// Criterion_64510408786520
// MI455X (gfx1250) — compile-verified
//
#include <hip/hip_runtime.h>
#include <math.h>

// Problem size (fixed by the reference).
constexpr int M_ROWS = 8192;
constexpr int N_COLS = 8192;
constexpr int TPB    = 256;                       // 8 wave32s per block
constexpr int F4_PER_ROW    = N_COLS / 4;         // 2048 float4 per row
constexpr int F4_PER_THREAD = F4_PER_ROW / TPB;   // 8 float4 per thread

#define LOG2E_F 1.4426950408889634f
#define LN2_F   0.6931471805599453f

// ---- CDNA5 async global->LDS path (gfx1250) -------------------------------
// Builtin prototype (from clang diagnostic): param0 is AS1 pointer to a
// 16-byte int vector, param1 is the AS3 (LDS) counterpart, then imm offset
// and imm cpol. Emits global_load_async_to_lds_b128 (ASYNCcnt-tracked).
#if defined(__gfx1250__) &&                                          \
    __has_builtin(__builtin_amdgcn_global_load_async_to_lds_b128) && \
    __has_builtin(__builtin_amdgcn_s_wait_asynccnt)
#define USE_ASYNC_LDS 1
typedef int v4i_t __attribute__((vector_size(16)));
typedef __attribute__((address_space(1))) v4i_t* gptr128_t;  // global
typedef __attribute__((address_space(3))) v4i_t* lptr128_t;  // LDS
#else
#define USE_ASYNC_LDS 0
#endif

// Kernel 1: one block per row. Computes lse[m] into row_out[m].
__global__ __launch_bounds__(TPB) void Criterion_row_lse_kernel(
    const float* __restrict__ x, const int* __restrict__ y,
    float* __restrict__ row_out)
{
    __shared__ float row[N_COLS];   // 32 KB staged row
    __shared__ float red_sq[TPB];
    __shared__ float red_s[TPB];

    const int r = blockIdx.x;
    const int t = threadIdx.x;

    const float4* gsrc = (const float4*)(x + (size_t)r * (size_t)N_COLS);
    float4* lrow = (float4*)row;

#if USE_ASYNC_LDS
    // Stream the whole row HBM -> LDS via the async data mover.
    // Per instruction: 32 lanes x 16B = 512B; 8 waves x 8 issues = 32 KB.
#pragma unroll
    for (int k = 0; k < F4_PER_THREAD; ++k) {
        const int i = t + k * TPB;
        __builtin_amdgcn_global_load_async_to_lds_b128(
            (gptr128_t)(gsrc + i), (lptr128_t)(lrow + i),
            /*offset=*/0, /*cpol=*/0);
    }
    __builtin_amdgcn_s_wait_asynccnt(0);
#else
#pragma unroll
    for (int k = 0; k < F4_PER_THREAD; ++k) {
        const int i = t + k * TPB;
        lrow[i] = gsrc[i];
    }
#endif
    __syncthreads();

    // Per-thread partials over a strided slice of the LDS row:
    //   sq = sum x^2              (fma chain)
    //   s  = sum exp2(2*log2e*x)  (= sum e^{2x}, native v_exp_f32)
    float sq = 0.0f;
    float s  = 0.0f;
#pragma unroll
    for (int k = 0; k < F4_PER_THREAD; ++k) {
        const float4 v = lrow[t + k * TPB];
        sq = fmaf(v.x, v.x, sq);
        sq = fmaf(v.y, v.y, sq);
        sq = fmaf(v.z, v.z, sq);
        sq = fmaf(v.w, v.w, sq);
        s += exp2f(v.x * (2.0f * LOG2E_F));
        s += exp2f(v.y * (2.0f * LOG2E_F));
        s += exp2f(v.z * (2.0f * LOG2E_F));
        s += exp2f(v.w * (2.0f * LOG2E_F));
    }

    red_sq[t] = sq;
    red_s[t]  = s;
    __syncthreads();
#pragma unroll
    for (int off = TPB / 2; off > 0; off >>= 1) {
        if (t < off) {
            red_sq[t] += red_sq[t + off];
            red_s[t]  += red_s[t + off];
        }
        __syncthreads();
    }

    if (t == 0) {
        const float sum_sq = red_sq[0];
        const float S      = red_s[0];          // sum_j exp2(t_j), t_j = 2*x_j*log2e
        const float xy     = row[y[r]];         // random access from staged LDS row
        const float C      = sum_sq + 1.0f;
        const float ty     = xy * (2.0f * LOG2E_F);
        // group j != y:  L1 = ln( S - e^{2 x_y} ) - C
        float S1 = S - exp2f(ty);
        S1 = fmaxf(S1, 1e-37f);                 // guard cancellation
        const float L1 = LN2_F * log2f(S1) - C;
        // the flipped y term: L2 = C - 2 x_y   (dominates for this data: C ~ N)
        const float L2 = C - 2.0f * xy;
        const float hi = fmaxf(L1, L2);
        const float lo = fminf(L1, L2);
        row_out[r] = hi + log1pf(expf(lo - hi));  // exact 2-term logaddexp
    }
}

// Kernel 2: deterministic mean of the M per-row lse values.
__global__ __launch_bounds__(TPB) void Criterion_mean_kernel(
    const float* __restrict__ v, float* __restrict__ out)
{
    __shared__ float red[TPB];
    float acc = 0.0f;
    for (int i = threadIdx.x; i < M_ROWS; i += TPB) acc += v[i];
    red[threadIdx.x] = acc;
    __syncthreads();
#pragma unroll
    for (int off = TPB / 2; off > 0; off >>= 1) {
        if (threadIdx.x < off) red[threadIdx.x] += red[threadIdx.x + off];
        __syncthreads();
    }
    if (threadIdx.x == 0) out[0] = red[0] * (1.0f / (float)M_ROWS);
}

extern "C" void kernel_launch(void* const* d_in, const int* in_sizes, int n_in,
                              void* d_out, int out_size, void* d_ws, size_t ws_size,
                              hipStream_t stream)
{
    const float* x = (const float*)d_in[0];
    const int*   y = (const int*)d_in[1];     // harness delivers integers as int32
    float* row_lse = (float*)d_ws;            // needs M_ROWS * 4 = 32 KB scratch
    float* out     = (float*)d_out;

    Criterion_row_lse_kernel<<<M_ROWS, TPB, 0, stream>>>(x, y, row_lse);
    Criterion_mean_kernel<<<1, TPB, 0, stream>>>(row_lse, out);
}